// GNN_38749194944748
// MI455X (gfx1250) — compile-verified
//
#include <hip/hip_runtime.h>
#include <hip/hip_bf16.h>

typedef __attribute__((ext_vector_type(2))) float v2f;
typedef __attribute__((ext_vector_type(8))) float v8f;

#define HDIM 32
#define SLOPE 0.2f

// ---------- monotone float <-> uint encoding for atomic segment-max ----------
__device__ __forceinline__ unsigned fenc(float f) {
    unsigned b = __float_as_uint(f);
    return (b & 0x80000000u) ? ~b : (b | 0x80000000u);
}
__device__ __forceinline__ float fdec(unsigned u) {
    unsigned b = (u & 0x80000000u) ? (u & 0x7FFFFFFFu) : ~u;
    return __uint_as_float(b);
}

// ---------- self-loop attr: sums & counts per destination ----------
__global__ void k_loop_accum(const int* __restrict__ dstI, const float* __restrict__ ea,
                             float* __restrict__ cnt, float* __restrict__ sums, int E) {
    int e = blockIdx.x * blockDim.x + threadIdx.x;
    if (e >= E) return;
    int d = dstI[e];
    atomicAdd(cnt + d, 1.0f);
    atomicAdd(sums + 4ull * d + 0, ea[4ull * e + 0]);
    atomicAdd(sums + 4ull * d + 1, ea[4ull * e + 1]);
    atomicAdd(sums + 4ull * d + 2, ea[4ull * e + 2]);
    atomicAdd(sums + 4ull * d + 3, ea[4ull * e + 3]);
}

__global__ void k_loop_div(const float* __restrict__ cnt, float* __restrict__ attr, int N) {
    int n = blockIdx.x * blockDim.x + threadIdx.x;
    if (n >= N) return;
    float c = fmaxf(cnt[n], 1.0f);
    attr[4ull * n + 0] /= c;
    attr[4ull * n + 1] /= c;
    attr[4ull * n + 2] /= c;
    attr[4ull * n + 3] /= c;
}

// ---------- c = We @ a_edge  (4 values) ----------
__global__ void k_cvec(const float* __restrict__ We, const float* __restrict__ a_edge,
                       float* __restrict__ c) {
    int k = threadIdx.x;
    if (k < 4) {
        float s = 0.f;
        #pragma unroll
        for (int j = 0; j < HDIM; ++j) s += We[k * HDIM + j] * a_edge[j];
        c[k] = s;
    }
}

// ---------- layer-1 transform: h = x@W1 (K=3), plus node scores ----------
__global__ void k_transform1(const float* __restrict__ x, const float* __restrict__ W1,
                             const float* __restrict__ a_src, const float* __restrict__ a_dst,
                             float* __restrict__ h, float* __restrict__ s_src,
                             float* __restrict__ s_dst, int N) {
    int n = blockIdx.x * blockDim.x + threadIdx.x;
    if (n >= N) return;
    float x0 = x[3ull * n], x1 = x[3ull * n + 1], x2 = x[3ull * n + 2];
    float ss = 0.f, sd = 0.f;
    #pragma unroll
    for (int j = 0; j < HDIM; ++j) {
        float hj = fmaf(x0, W1[j], fmaf(x1, W1[HDIM + j], x2 * W1[2 * HDIM + j]));
        h[(size_t)n * HDIM + j] = hj;
        ss = fmaf(hj, a_src[j], ss);
        sd = fmaf(hj, a_dst[j], sd);
    }
    s_src[n] = ss;
    s_dst[n] = sd;
}

// ---------- layer-2 transform via fp32 WMMA: H = A(N x 32) @ W(32 x 32) ----------
// One wave per 16-row tile; two 16-col halves; K=32 in 8 steps of 4.
__global__ void k_transform2_wmma(const float* __restrict__ A, const float* __restrict__ W,
                                  float* __restrict__ Hout, int nTiles) {
    int wave = (int)((blockIdx.x * (size_t)blockDim.x + threadIdx.x) >> 5);
    if (wave >= nTiles) return;                    // wave-uniform: EXEC stays all-1s
    int lane = threadIdx.x & 31;
    int m    = lane & 15;                          // A row (within tile) / B column
    int hi   = lane >> 4;                          // lane-half selects K pair / M half
    const float* Arow = A + (size_t)(wave * 16 + m) * HDIM;

    #pragma unroll
    for (int nt = 0; nt < 2; ++nt) {
        v8f c = {};
        #pragma unroll
        for (int k0 = 0; k0 < HDIM; k0 += 4) {
            v2f a, b;
            // A 16x4 layout: lanes 0-15 -> K = k0,k0+1 ; lanes 16-31 -> K = k0+2,k0+3
            a.x = Arow[k0 + 2 * hi];
            a.y = Arow[k0 + 2 * hi + 1];
            // B 4x16 layout (mirrored): VGPR0 = row k0(+2), VGPR1 = row k0+1(+3)
            int nc = nt * 16 + m;
            b.x = W[(size_t)(k0 + 2 * hi) * HDIM + nc];
            b.y = W[(size_t)(k0 + 2 * hi + 1) * HDIM + nc];
            c = __builtin_amdgcn_wmma_f32_16x16x4_f32(
                    false, a, false, b, (short)0, c, false, false);
        }
        // D 16x16 layout: VGPR v -> M = v + 8*hi, N = lane&15
        #pragma unroll
        for (int v = 0; v < 8; ++v) {
            int Mrow = v + 8 * hi;
            Hout[(size_t)(wave * 16 + Mrow) * HDIM + nt * 16 + m] = c[v];
        }
    }
}

// ---------- per-node attention scores (wave per node, butterfly reduce) ----------
__global__ void k_node_scores(const float* __restrict__ h, const float* __restrict__ a_src,
                              const float* __restrict__ a_dst, float* __restrict__ s_src,
                              float* __restrict__ s_dst, int N) {
    int wave = (int)((blockIdx.x * (size_t)blockDim.x + threadIdx.x) >> 5);
    if (wave >= N) return;
    int lane = threadIdx.x & 31;
    float hv = h[(size_t)wave * HDIM + lane];
    float ts = hv * a_src[lane];
    float td = hv * a_dst[lane];
    #pragma unroll
    for (int o = 16; o > 0; o >>= 1) {
        ts += __shfl_down(ts, o, 32);
        td += __shfl_down(td, o, 32);
    }
    if (lane == 0) { s_src[wave] = ts; s_dst[wave] = td; }
}

// ---------- edge logits + segment max (thread per extended edge) ----------
__global__ void k_logits_max(const int* __restrict__ srcI, const int* __restrict__ dstI,
                             const float* __restrict__ ea, const float* __restrict__ loop_attr,
                             const float* __restrict__ cvec, const float* __restrict__ s_src,
                             const float* __restrict__ s_dst, float* __restrict__ logits,
                             unsigned* __restrict__ mseg, int E, int EX) {
    int e = blockIdx.x * blockDim.x + threadIdx.x;
    if (e >= EX) return;
    int s, d;
    const float* a;
    if (e < E) { s = srcI[e]; d = dstI[e]; a = ea + 4ull * e; }
    else       { s = d = e - E;            a = loop_attr + 4ull * (e - E); }
    float dot = fmaf(a[0], cvec[0], fmaf(a[1], cvec[1], fmaf(a[2], cvec[2], a[3] * cvec[3])));
    float l = s_src[s] + s_dst[d] + dot;
    l = (l > 0.f) ? l : SLOPE * l;
    logits[e] = l;
    atomicMax(mseg + d, fenc(l));
}

// ---------- exp + unnormalized aggregation (wave per extended edge) ----------
__global__ void k_exp_agg(const int* __restrict__ srcI, const int* __restrict__ dstI,
                          const float* __restrict__ logits, const unsigned* __restrict__ mseg,
                          float* __restrict__ denom, const float* __restrict__ h,
                          float* __restrict__ agg, int E, int EX) {
    int wave = (int)((blockIdx.x * (size_t)blockDim.x + threadIdx.x) >> 5);
    if (wave >= EX) return;
    int lane = threadIdx.x & 31;
    int s, d;
    if (wave < E) { s = srcI[wave]; d = dstI[wave]; }
    else          { s = d = wave - E; }
    float w = __expf(logits[wave] - fdec(mseg[d]));
    if (lane == 0) atomicAdd(denom + d, w);
    float hv = h[(size_t)s * HDIM + lane];
    atomicAdd(agg + (size_t)d * HDIM + lane, w * hv);
}

// ---------- normalize + bias (+relu) -> next-layer input ----------
__global__ void k_finish(const float* __restrict__ agg, const float* __restrict__ denom,
                         const float* __restrict__ bias, float* __restrict__ outb,
                         int N, int do_relu) {
    int wave = (int)((blockIdx.x * (size_t)blockDim.x + threadIdx.x) >> 5);
    if (wave >= N) return;
    int lane = threadIdx.x & 31;
    float v = agg[(size_t)wave * HDIM + lane] / denom[wave] + bias[lane];
    if (do_relu) v = fmaxf(v, 0.f);
    outb[(size_t)wave * HDIM + lane] = v;
}

// ---------- layer-2 finish fused with final Linear(32->1) + relu ----------
__global__ void k_final(const float* __restrict__ agg, const float* __restrict__ denom,
                        const float* __restrict__ b2, const float* __restrict__ Wl,
                        const float* __restrict__ bl, float* __restrict__ out, int N) {
    int wave = (int)((blockIdx.x * (size_t)blockDim.x + threadIdx.x) >> 5);
    if (wave >= N) return;
    int lane = threadIdx.x & 31;
    float v = agg[(size_t)wave * HDIM + lane] / denom[wave] + b2[lane];
    float t = v * Wl[lane];
    #pragma unroll
    for (int o = 16; o > 0; o >>= 1) t += __shfl_down(t, o, 32);
    if (lane == 0) out[wave] = fmaxf(t + bl[0], 0.f);
}

extern "C" void kernel_launch(void* const* d_in, const int* in_sizes, int n_in,
                              void* d_out, int out_size, void* d_ws, size_t ws_size,
                              hipStream_t stream) {
    const float* x        = (const float*)d_in[0];
    const int*   eidx     = (const int*)  d_in[1];
    const float* ea       = (const float*)d_in[2];
    const float* W1       = (const float*)d_in[3];
    const float* We1      = (const float*)d_in[4];
    const float* a_src1   = (const float*)d_in[5];
    const float* a_dst1   = (const float*)d_in[6];
    const float* a_edge1  = (const float*)d_in[7];
    const float* b1       = (const float*)d_in[8];
    const float* W2       = (const float*)d_in[9];
    const float* We2      = (const float*)d_in[10];
    const float* a_src2   = (const float*)d_in[11];
    const float* a_dst2   = (const float*)d_in[12];
    const float* a_edge2  = (const float*)d_in[13];
    const float* b2       = (const float*)d_in[14];
    const float* Wl       = (const float*)d_in[15];
    const float* bl       = (const float*)d_in[16];

    const int N  = in_sizes[0] / 3;
    const int E  = in_sizes[1] / 2;
    const int EX = E + N;
    const int* srcI = eidx;       // edge_index row 0
    const int* dstI = eidx + E;   // edge_index row 1

    // ---- workspace layout (floats) ----
    float* ws = (float*)d_ws;
    float*    loop_cnt  = ws;                         // N
    float*    loop_attr = ws + (size_t)N;             // 4N  (sums -> means in place)
    unsigned* mseg      = (unsigned*)(ws + 5ull * N); // N
    float*    denom     = ws + 6ull  * N;             // N
    float*    agg       = ws + 7ull  * N;             // 32N
    float*    h         = ws + 39ull * N;             // 32N  (transformed features)
    float*    s_src     = ws + 71ull * N;             // N
    float*    s_dst     = ws + 72ull * N;             // N
    float*    hbuf      = ws + 73ull * N;             // 32N  (relu(layer1 out))
    float*    logits    = ws + 105ull * N;            // EX
    float*    cvec      = logits + (size_t)EX;        // 8

    const int TPB = 256;
    dim3 blk(TPB);
    auto cdiv = [](long long a, long long b) { return (int)((a + b - 1) / b); };
    int gN      = cdiv(N, TPB);
    int gE      = cdiv(E, TPB);
    int gEX     = cdiv(EX, TPB);
    int gWaveN  = cdiv((long long)N * 32, TPB);
    int gWaveEX = cdiv((long long)EX * 32, TPB);
    int nTiles  = (N + 15) / 16;
    int gWMMA   = cdiv((long long)nTiles * 32, TPB);

    // ---- self-loop edge attr (mean of incoming edge_attr per dst) ----
    hipMemsetAsync(loop_cnt, 0, 5ull * N * sizeof(float), stream);
    k_loop_accum<<<gE, blk, 0, stream>>>(dstI, ea, loop_cnt, loop_attr, E);
    k_loop_div<<<gN, blk, 0, stream>>>(loop_cnt, loop_attr, N);

    // ================= layer 1 =================
    k_transform1<<<gN, blk, 0, stream>>>(x, W1, a_src1, a_dst1, h, s_src, s_dst, N);
    k_cvec<<<1, 32, 0, stream>>>(We1, a_edge1, cvec);
    hipMemsetAsync(mseg, 0, 34ull * N * sizeof(float), stream);  // mseg+denom+agg
    k_logits_max<<<gEX, blk, 0, stream>>>(srcI, dstI, ea, loop_attr, cvec,
                                          s_src, s_dst, logits, mseg, E, EX);
    k_exp_agg<<<gWaveEX, blk, 0, stream>>>(srcI, dstI, logits, mseg, denom, h, agg, E, EX);
    k_finish<<<gWaveN, blk, 0, stream>>>(agg, denom, b1, hbuf, N, /*relu=*/1);

    // ================= layer 2 =================
    k_transform2_wmma<<<gWMMA, blk, 0, stream>>>(hbuf, W2, h, nTiles);
    k_node_scores<<<gWaveN, blk, 0, stream>>>(h, a_src2, a_dst2, s_src, s_dst, N);
    k_cvec<<<1, 32, 0, stream>>>(We2, a_edge2, cvec);
    hipMemsetAsync(mseg, 0, 34ull * N * sizeof(float), stream);
    k_logits_max<<<gEX, blk, 0, stream>>>(srcI, dstI, ea, loop_attr, cvec,
                                          s_src, s_dst, logits, mseg, E, EX);
    k_exp_agg<<<gWaveEX, blk, 0, stream>>>(srcI, dstI, logits, mseg, denom, h, agg, E, EX);

    // ---- final Linear(32->1) + relu, fused with layer-2 normalize ----
    k_final<<<gWaveN, blk, 0, stream>>>(agg, denom, b2, Wl, bl, (float*)d_out, N);
}